// Model_69767448756492
// MI455X (gfx1250) — compile-verified
//
#include <hip/hip_runtime.h>
#include <hip/hip_bf16.h>
#include <stdint.h>

// Scatter-add with sorted indices: out[idx[i], :] += value[i, :]
//   N = 262144 output rows, M = 1048576 value rows, D = 128 (float32)
// Memory-bound: ~775 MB of single-touch traffic -> ~33 us at 23.3 TB/s HBM.
// Strategy:
//   - segment boundaries from the sorted index (no atomics -> deterministic)
//   - one wave32 per output row; lane l owns float4 columns [4l, 4l+4)
//   - value rows streamed via CDNA5 async global->LDS DMA (ASYNCcnt) with
//     th:TH_LOAD_NT (one-touch stream: do not pollute the 192 MB L2)
//   - self/out via non-temporal b128 loads/stores (also one-touch)

#define DCOLS 128              // floats per row
#define V4_PER_ROW 32          // float4 per row == lanes per wave
#define WAVES_PER_WG 8         // 256 threads = 8 wave32
#define BATCH 8                // value rows staged per async batch

typedef float v4f __attribute__((ext_vector_type(4)));

// ---------------- Kernel A: mark segment boundaries --------------------
__global__ __launch_bounds__(256) void build_segments(
    const int* __restrict__ idx, unsigned* __restrict__ seg_start,
    unsigned* __restrict__ seg_end, int M)
{
    int i = blockIdx.x * blockDim.x + threadIdx.x;
    if (i >= M) return;
    int n = idx[i];
    if (i == 0 || idx[i - 1] != n) seg_start[n] = (unsigned)i;
    if (i == M - 1 || idx[i + 1] != n) seg_end[n] = (unsigned)(i + 1);
}

// ---------------- Kernel B: per-row segment sum ------------------------
__global__ __launch_bounds__(256) void scatter_add_rows(
    const float* __restrict__ self, const float* __restrict__ value,
    const unsigned* __restrict__ seg_start, const unsigned* __restrict__ seg_end,
    float* __restrict__ out, int N)
{
    // 8 waves * 8 rows * 128 floats * 4B = 32 KB LDS per workgroup
    __shared__ float smem[WAVES_PER_WG * BATCH * DCOLS];

    const int wave = threadIdx.x >> 5;
    const int lane = threadIdx.x & 31;
    const int n = blockIdx.x * WAVES_PER_WG + wave;   // wave-uniform row id
    if (n >= N) return;                                // uniform exit per wave

    // accumulator starts from self row (non-temporal: read exactly once)
    const v4f* self4 = (const v4f*)self;
    v4f acc = __builtin_nontemporal_load(&self4[(size_t)n * V4_PER_ROW + lane]);

    const unsigned lo = seg_start[n];
    const unsigned hi = seg_end[n];

    // raw LDS byte offset of this wave's staging buffer
    // (generic AS3 pointer: low 32 bits == LDS offset on gfx1250)
    float* mybuf = &smem[wave * (BATCH * DCOLS)];
    const unsigned lds_base = (unsigned)(uintptr_t)mybuf;

    // per-lane global pointer into the first row of the segment
    const char* grow = (const char*)value +
        ((size_t)lo * DCOLS + (unsigned)lane * 4u) * sizeof(float);

    for (unsigned base = lo; base < hi; base += BATCH) {
        unsigned cnt = hi - base;
        if (cnt > BATCH) cnt = BATCH;

        // issue async DMA: each lane moves 16B of row (base+j) into LDS;
        // NT hint: 512 MB one-touch stream must not thrash L2
        for (unsigned j = 0; j < cnt; ++j) {
            unsigned laddr = lds_base + j * (DCOLS * 4u) + (unsigned)lane * 16u;
            asm volatile("global_load_async_to_lds_b128 %0, %1, off th:TH_LOAD_NT"
                         :: "v"(laddr), "v"(grow)
                         : "memory");
            grow += DCOLS * sizeof(float);
        }
        // wait for all async transfers of this batch to land in LDS
        asm volatile("s_wait_asynccnt 0x0" ::: "memory");

        // accumulate in fixed order -> bitwise deterministic across calls
        const v4f* sm4 = (const v4f*)mybuf;
        for (unsigned j = 0; j < cnt; ++j)
            acc += sm4[j * V4_PER_ROW + lane];
    }

    v4f* out4 = (v4f*)out;
    __builtin_nontemporal_store(acc, &out4[(size_t)n * V4_PER_ROW + lane]);
}

// ---------------- host-side launcher -----------------------------------
extern "C" void kernel_launch(void* const* d_in, const int* in_sizes, int n_in,
                              void* d_out, int out_size, void* d_ws, size_t ws_size,
                              hipStream_t stream)
{
    const float* self  = (const float*)d_in[0];   // (N, 128) f32
    const float* value = (const float*)d_in[1];   // (M, 128) f32
    const int*   idx   = (const int*)d_in[2];     // (M,) int32 (sorted)
    // d_in[3] = pos, unused by the reference.

    const int N = in_sizes[0] / DCOLS;
    const int M = in_sizes[2];

    unsigned* seg_start = (unsigned*)d_ws;
    unsigned* seg_end   = seg_start + N;

    // zero segment tables (rows with no contributions -> lo == hi == 0)
    hipMemsetAsync(d_ws, 0, (size_t)2 * (size_t)N * sizeof(unsigned), stream);

    build_segments<<<(M + 255) / 256, 256, 0, stream>>>(idx, seg_start, seg_end, M);

    scatter_add_rows<<<(N + WAVES_PER_WG - 1) / WAVES_PER_WG, 256, 0, stream>>>(
        self, value, seg_start, seg_end, (float*)d_out, N);
}